// PaiConv_35064113005153
// MI455X (gfx1250) — compile-verified
//
#include <hip/hip_runtime.h>
#include <math.h>

#define BB   16
#define NPTS 2048
#define KNN  20
#define INC  64
#define OUTC 64
#define NF   192

typedef float v2f __attribute__((ext_vector_type(2)));
typedef float v8f __attribute__((ext_vector_type(8)));

// ---- order-preserving float <-> uint mapping for atomic min/max ----
__device__ __forceinline__ unsigned fmap(float f) {
    unsigned u = __float_as_uint(f);
    return (u & 0x80000000u) ? ~u : (u | 0x80000000u);
}
__device__ __forceinline__ float funmap(unsigned k) {
    return (k & 0x80000000u) ? __uint_as_float(k & 0x7FFFFFFFu)
                             : __uint_as_float(~k);
}

// =====================================================================
// Kernel 1: kNN top-20 (smallest squared distance, stable ties)
// grid 128 = B * (N/256), block 256. Point cloud + sq-norms in LDS.
// =====================================================================
__global__ __launch_bounds__(256) void knn_kernel(const float* __restrict__ x,
                                                  int* __restrict__ idxws) {
    __shared__ float sp[NPTS * 3];
    __shared__ float ssq[NPTS];
    const int b = blockIdx.x >> 3;
    const int chunk = blockIdx.x & 7;
    const float* xb = x + (size_t)b * 3 * NPTS;
    for (int i = threadIdx.x; i < NPTS; i += 256) {
        float p0 = xb[i], p1 = xb[NPTS + i], p2 = xb[2 * NPTS + i];
        sp[i * 3 + 0] = p0; sp[i * 3 + 1] = p1; sp[i * 3 + 2] = p2;
        ssq[i] = p0 * p0 + p1 * p1 + p2 * p2;
    }
    __syncthreads();
    const int q = chunk * 256 + threadIdx.x;
    const float qx = sp[q * 3], qy = sp[q * 3 + 1], qz = sp[q * 3 + 2];
    const float sqq = ssq[q];
    float bd[KNN];
    int   bi[KNN];
#pragma unroll
    for (int j = 0; j < KNN; ++j) { bd[j] = 3.4e38f; bi[j] = 0; }
    for (int m = 0; m < NPTS; ++m) {
        float d = sqq - 2.0f * (qx * sp[m * 3] + qy * sp[m * 3 + 1] + qz * sp[m * 3 + 2]) + ssq[m];
        if (d < bd[KNN - 1]) {
            float cd = d; int ci = m;
#pragma unroll
            for (int j = 0; j < KNN; ++j) {
                if (cd < bd[j]) {
                    float td = bd[j]; bd[j] = cd; cd = td;
                    int ti = bi[j]; bi[j] = ci; ci = ti;
                }
            }
        }
    }
    int* op = idxws + ((size_t)b * NPTS + q) * KNN;
#pragma unroll
    for (int j = 0; j < KNN; ++j) op[j] = bi[j];
}

// =====================================================================
// Kernel 2a/2b: global min/max of x_feats entries (rep, rel, dis)
// =====================================================================
__global__ void minmax_init_kernel(unsigned* mm) {
    mm[0] = 0xFFFFFFFFu;  // min key
    mm[1] = 0u;           // max key
}

__global__ __launch_bounds__(256) void minmax_kernel(const float* __restrict__ x,
                                                     const int* __restrict__ idxws,
                                                     unsigned* __restrict__ mm) {
    const int t = blockIdx.x * 256 + threadIdx.x;   // over B*N
    const int b = t >> 11, n = t & (NPTS - 1);
    const float* xb = x + (size_t)b * 3 * NPTS;
    const int* ip = idxws + ((size_t)b * NPTS + n) * KNN;
    const int i0 = ip[0];
    const float r0 = xb[i0], r1 = xb[NPTS + i0], r2 = xb[2 * NPTS + i0];
    float mn = fminf(fminf(r0, r1), r2);
    float mx = fmaxf(fmaxf(r0, r1), r2);
    for (int k = 0; k < KNN; ++k) {
        int id = ip[k];
        float e0 = xb[id] - r0, e1 = xb[NPTS + id] - r1, e2 = xb[2 * NPTS + id] - r2;
        float dis = sqrtf(e0 * e0 + e1 * e1 + e2 * e2);
        mn = fminf(mn, fminf(fminf(e0, e1), fminf(e2, dis)));
        mx = fmaxf(mx, fmaxf(fmaxf(e0, e1), fmaxf(e2, dis)));
    }
#pragma unroll
    for (int off = 1; off < 32; off <<= 1) {
        mn = fminf(mn, __shfl_xor(mn, off, 32));
        mx = fmaxf(mx, __shfl_xor(mx, off, 32));
    }
    if ((threadIdx.x & 31) == 0) {
        atomicMin(&mm[0], fmap(mn));
        atomicMax(&mm[1], fmap(mx));
    }
}

// =====================================================================
// Kernel 3: fused per-point pipeline. 1 wave per point, 4 waves/block.
// =====================================================================
struct WaveScratch {
    int   idx[KNN];
    float nb[KNN][3];
    float perm[KNN * 8];
    float S[8];
    float xmlp[32 * KNN];   // [o][k]
    float res[OUTC];
    float agg[NF * 8];      // original channel order
};

__global__ __launch_bounds__(128) void paiconv_main(
    const float* __restrict__ x, const float* __restrict__ feature,
    const int* __restrict__ idxws, const unsigned* __restrict__ mm,
    const float* __restrict__ Bf, const float* __restrict__ kern,
    const float* __restrict__ A, const float* __restrict__ pad,
    const float* __restrict__ Wmlp, const float* __restrict__ bmlp,
    const float* __restrict__ W1, const float* __restrict__ b1,
    const float* __restrict__ Wout, float* __restrict__ dout) {
    __shared__ WaveScratch wsc[4];
    __shared__ float sWmlp[32 * 64];
    __shared__ float sBf[7 * 32];
    __shared__ float sM[3 * 8];
    __shared__ float sPad[KNN * 8];
    __shared__ float sBmlp[32];

    const int tid = threadIdx.x;
    const int lane = tid & 31;
    const int w = tid >> 5;
    const int point = blockIdx.x * 4 + w;
    const int b = point >> 11;
    const int n = point & (NPTS - 1);

    // ---- phase 0: stage small constants block-wide ----
    for (int i = tid; i < 32 * 64; i += 128) sWmlp[i] = Wmlp[i];
    for (int i = tid; i < 7 * 32; i += 128) sBf[i] = Bf[i];
    for (int i = tid; i < KNN * 8; i += 128) sPad[i] = pad[i];
    if (tid < 32) sBmlp[tid] = bmlp[tid];
    if (tid < 24) {
        int d = tid / 8, e = tid % 8;
        float acc = 0.f;
        for (int g = 0; g < 3; ++g)
            acc += 0.5f * (A[d * 3 + g] + A[g * 3 + d]) * kern[g * 8 + e];
        sM[tid] = acc;
    }
    __syncthreads();

    WaveScratch& Sv = wsc[w];

    // ---- phase 1: load neighbor indices + coords ----
    if (lane < KNN) {
        int id = idxws[((size_t)b * NPTS + n) * KNN + lane];
        Sv.idx[lane] = id;
        const float* xb = x + (size_t)b * 3 * NPTS;
        Sv.nb[lane][0] = xb[id];
        Sv.nb[lane][1] = xb[NPTS + id];
        Sv.nb[lane][2] = xb[2 * NPTS + id];
    }
    __syncthreads();

    // ---- phase 2: Fourier features -> sin/cos -> x_mlp; perm logits ----
    if (lane < KNN) {
        const int k = lane;
        float r0 = Sv.nb[0][0], r1 = Sv.nb[0][1], r2 = Sv.nb[0][2];
        float e0 = Sv.nb[k][0] - r0, e1 = Sv.nb[k][1] - r1, e2 = Sv.nb[k][2] - r2;
        float dis = sqrtf(e0 * e0 + e1 * e1 + e2 * e2);
        float mn = funmap(mm[0]), mx = funmap(mm[1]);
        float off = mn / (mx - mn);
        float t[7] = {r0, r1, r2, e0, e1, e2, dis};
#pragma unroll
        for (int i = 0; i < 7; ++i) t[i] = 6.2831853071795865f * (t[i] - off);
        float sn[32], cs[32];
#pragma unroll
        for (int j = 0; j < 32; ++j) {
            float a = 0.f;
#pragma unroll
            for (int i = 0; i < 7; ++i) a += t[i] * sBf[i * 32 + j];
            sn[j] = sinf(a);
            cs[j] = cosf(a);
        }
#pragma unroll 4
        for (int o = 0; o < 32; ++o) {
            float a = sBmlp[o];
#pragma unroll
            for (int j = 0; j < 32; ++j)
                a += sn[j] * sWmlp[o * 64 + j] + cs[j] * sWmlp[o * 64 + 32 + j];
            Sv.xmlp[o * KNN + k] = a;
        }
#pragma unroll
        for (int e = 0; e < 8; ++e)
            Sv.perm[k * 8 + e] = sPad[k * 8 + e] + e0 * sM[e] + e1 * sM[8 + e] + e2 * sM[16 + e];
    }
    __syncthreads();

    // ---- phase 3: softmax over k per column e, threshold > 1/K, renorm ----
    if (lane < 8) {
        const int e = lane;
        float mxl = -3.4e38f;
        for (int k = 0; k < KNN; ++k) mxl = fmaxf(mxl, Sv.perm[k * 8 + e]);
        float p[KNN]; float sum = 0.f;
        for (int k = 0; k < KNN; ++k) { p[k] = expf(Sv.perm[k * 8 + e] - mxl); sum += p[k]; }
        float inv = 1.f / sum; float s2 = 0.f;
        for (int k = 0; k < KNN; ++k) {
            float v = p[k] * inv;
            v = (v > (1.0f / (float)KNN)) ? v : 0.f;
            p[k] = v; s2 += v;
        }
        float invs = 1.f / (s2 + 1e-6f); float Ssum = 0.f;
        for (int k = 0; k < KNN; ++k) {
            float v = p[k] * invs;
            Sv.perm[k * 8 + e] = v; Ssum += v;
        }
        Sv.S[e] = Ssum;
    }
    __syncthreads();

    // ---- phase 4: G = feats0 * perm, build agg (orig channel order); residual ----
    {
        const float* fb = feature + (size_t)b * INC * NPTS;
        const int c1 = lane, c2 = lane + 32;
        float a1[8], a2[8], a3[8];
#pragma unroll
        for (int m = 0; m < 8; ++m) { a1[m] = 0.f; a2[m] = 0.f; a3[m] = 0.f; }
        float f1r = 0.f, f2r = 0.f, f3r = 0.f;
        for (int k = 0; k < KNN; ++k) {
            int id = Sv.idx[k];
            float f1 = fb[c1 * NPTS + id];
            float f2 = fb[c2 * NPTS + id];
            float f3 = Sv.xmlp[lane * KNN + k];
            if (k == 0) { f1r = f1; f2r = f2; f3r = f3; }
#pragma unroll
            for (int m = 0; m < 8; ++m) {
                float p = Sv.perm[k * 8 + m];
                a1[m] += f1 * p; a2[m] += f2 * p; a3[m] += f3 * p;
            }
        }
#pragma unroll
        for (int m = 0; m < 8; ++m) {
            float sm = Sv.S[m];
            Sv.agg[c1 * 8 + m]          = f1r * sm;
            Sv.agg[(c1 + 96) * 8 + m]   = a1[m] - f1r * sm;
            Sv.agg[c2 * 8 + m]          = f2r * sm;
            Sv.agg[(c2 + 96) * 8 + m]   = a2[m] - f2r * sm;
            Sv.agg[(lane + 64) * 8 + m]        = f3r * sm;
            Sv.agg[(lane + 64 + 96) * 8 + m]   = a3[m] - f3r * sm;
        }
        // residual: feature[b,:,n] @ W_out^T  (channels lane, lane+32)
        float racc0 = 0.f, racc1 = 0.f;
        for (int c = 0; c < INC; ++c) {
            float fc = fb[c * NPTS + n];
            racc0 += fc * Wout[lane * INC + c];
            racc1 += fc * Wout[(lane + 32) * INC + c];
        }
        Sv.res[lane] = racc0;
        Sv.res[lane + 32] = racc1;
    }
    __syncthreads();

    // ---- phase 5: W1(64x192) x agg(192x8) via V_WMMA_F32_16X16X4_F32 ----
    // Iterate W1 columns c' contiguously; the group shuffle maps W1 column
    // block {4kk..4kk+3} to agg rows {kk, 48+kk, 96+kk, 144+kk}: constant
    // strides, contiguous b64 A-loads, unconditional LDS B-loads.
    {
        const int cn = lane & 15;                  // A row (o), B col (m)
        const int ksel = (lane < 16) ? 0 : 2;      // K rows handled by this half
        const int col = cn & 7;                    // clamped B column (broadcast dup)
        const bool valid = (cn < 8);
        const float* aggR0 = &Sv.agg[(ksel * 48) * 8 + col];       // + kk*8
        const float* aggR1 = &Sv.agg[((ksel + 1) * 48) * 8 + col]; // + kk*8
#pragma unroll
        for (int ot = 0; ot < 4; ++ot) {
            const float* wrow = W1 + (16 * ot + cn) * NF + ksel;   // + 4*kk
            v8f acc = {};
#pragma unroll 4
            for (int kk = 0; kk < 48; ++kk) {
                v2f av = *(const v2f*)(wrow + 4 * kk);
                float bx = aggR0[kk * 8];
                float by = aggR1[kk * 8];
                v2f bv;
                bv.x = valid ? bx : 0.f;
                bv.y = valid ? by : 0.f;
                acc = __builtin_amdgcn_wmma_f32_16x16x4_f32(
                    false, av, false, bv, (short)0, acc, false, false);
            }
            // max over m (cols 0..7 per half-wave), + b1, exact gelu, + residual
#pragma unroll
            for (int r = 0; r < 8; ++r) {
                float v = acc[r];
                v = fmaxf(v, __shfl_xor(v, 1, 32));
                v = fmaxf(v, __shfl_xor(v, 2, 32));
                v = fmaxf(v, __shfl_xor(v, 4, 32));
                if ((lane & 15) == 0) {
                    int o = 16 * ot + r + ((lane == 16) ? 8 : 0);
                    float h = v + b1[o];
                    float g = 0.5f * h * (1.0f + erff(h * 0.7071067811865475f));
                    dout[(size_t)b * OUTC * NPTS + (size_t)o * NPTS + n] = g + Sv.res[o];
                }
            }
        }
    }
}

// =====================================================================
// Kernel 4: per-channel mean/var over (B, N)
// =====================================================================
__global__ __launch_bounds__(256) void stats_kernel(const float* __restrict__ h,
                                                    float* __restrict__ stats) {
    const int o = blockIdx.x;
    float s = 0.f, q = 0.f;
    for (int i = threadIdx.x; i < BB * NPTS; i += 256) {
        int b = i >> 11, n = i & (NPTS - 1);
        float v = h[(size_t)b * OUTC * NPTS + (size_t)o * NPTS + n];
        s += v; q += v * v;
    }
    __shared__ float ss[256], sq2[256];
    ss[threadIdx.x] = s; sq2[threadIdx.x] = q;
    __syncthreads();
    for (int off = 128; off > 0; off >>= 1) {
        if (threadIdx.x < off) {
            ss[threadIdx.x] += ss[threadIdx.x + off];
            sq2[threadIdx.x] += sq2[threadIdx.x + off];
        }
        __syncthreads();
    }
    if (threadIdx.x == 0) {
        float mu = ss[0] / (float)(BB * NPTS);
        float var = sq2[0] / (float)(BB * NPTS) - mu * mu;
        stats[o] = mu;
        stats[OUTC + o] = var;
    }
}

// =====================================================================
// Kernel 5: in-place normalization of d_out
// =====================================================================
__global__ __launch_bounds__(256) void norm_kernel(float* __restrict__ h,
                                                   const float* __restrict__ stats,
                                                   const float* __restrict__ gamma,
                                                   const float* __restrict__ beta) {
    const size_t t = (size_t)blockIdx.x * 256 + threadIdx.x;
    const int o = (int)((t >> 11) & 63);
    float mu = stats[o], var = stats[OUTC + o];
    h[t] = gamma[o] * (h[t] - mu) * rsqrtf(var + 1e-5f) + beta[o];
}

// =====================================================================
extern "C" void kernel_launch(void* const* d_in, const int* in_sizes, int n_in,
                              void* d_out, int out_size, void* d_ws, size_t ws_size,
                              hipStream_t stream) {
    (void)in_sizes; (void)n_in; (void)out_size; (void)ws_size;
    const float* x       = (const float*)d_in[0];
    const float* feature = (const float*)d_in[1];
    const float* A       = (const float*)d_in[2];
    const float* Bf      = (const float*)d_in[3];
    const float* kern    = (const float*)d_in[4];
    const float* pad     = (const float*)d_in[5];
    const float* Wmlp    = (const float*)d_in[6];
    const float* bmlp    = (const float*)d_in[7];
    const float* W1      = (const float*)d_in[8];
    const float* b1      = (const float*)d_in[9];
    const float* gamma   = (const float*)d_in[10];
    const float* beta    = (const float*)d_in[11];
    const float* Wout    = (const float*)d_in[12];
    float* out = (float*)d_out;

    char* wsb = (char*)d_ws;
    int* idxws = (int*)wsb;                                 // B*N*K ints = 2,621,440 B
    unsigned* mm = (unsigned*)(wsb + 2621440);              // 8 B
    float* stats = (float*)(wsb + 2621504);                 // 512 B

    knn_kernel<<<BB * (NPTS / 256), 256, 0, stream>>>(x, idxws);
    minmax_init_kernel<<<1, 1, 0, stream>>>(mm);
    minmax_kernel<<<(BB * NPTS) / 256, 256, 0, stream>>>(x, idxws, mm);
    paiconv_main<<<(BB * NPTS) / 4, 128, 0, stream>>>(
        x, feature, idxws, mm, Bf, kern, A, pad, Wmlp, bmlp, W1, b1, Wout, out);
    stats_kernel<<<OUTC, 256, 0, stream>>>(out, stats);
    norm_kernel<<<(BB * OUTC * NPTS) / 256, 256, 0, stream>>>(out, stats, gamma, beta);
}